// GATEncoder_5677946765450
// MI455X (gfx1250) — compile-verified
//
#include <hip/hip_runtime.h>
#include <math.h>

// ---------------------------------------------------------------------------
// GAT encoder (3 layers) for MI455X / gfx1250.
// Dense projections via V_WMMA_F32_16X16X4_F32 (full f32 precision).
// Edge softmax + aggregation via coalesced per-edge kernels with f32 atomics.
// ---------------------------------------------------------------------------

typedef __attribute__((ext_vector_type(2))) float v2f;
typedef __attribute__((ext_vector_type(8))) float v8f;

#define F_IN 128
#define HIDC 128   // hidden channels (= HEADS*C for concat layers, = OUT)
#define EPS_BN 1e-5f

static inline int ceil_div(int a, int b) { return (a + b - 1) / b; }

// ------------------------- WMMA f32 GEMM -----------------------------------
// C[M,Ncols] = A[M,K] @ B[K,Ncols(ldb)]   -- one wave per 16x16 output tile,
// K accumulated in 32 chained 16x16x4 f32 WMMAs. M%16==0, Ncols%16==0, K%4==0.
__global__ __launch_bounds__(256) void wmma_gemm_f32(
    const float* __restrict__ A, int lda,
    const float* __restrict__ B, int ldb,
    float* __restrict__ C, int ldc,
    int M, int Ncols, int K)
{
    const int lane = threadIdx.x & 31;
    const int wave = threadIdx.x >> 5;
    const int tilesN = Ncols >> 4;
    const int tilesM = M >> 4;
    const int tile = blockIdx.x * (blockDim.x >> 5) + wave;
    if (tile >= tilesM * tilesN) return;   // wave-uniform: EXEC stays all-ones

    const int tm = tile / tilesN;
    const int tn = tile - tm * tilesN;
    const int m0 = tm << 4;
    const int n0 = tn << 4;

    const int half = lane >> 4;   // 0: K pair (k0,k0+1), 1: (k0+2,k0+3)
    const int l    = lane & 15;   // M index (A) / N index (B,D)
    const float* Arow = A + (size_t)(m0 + l) * lda;

    v8f acc = {};
    for (int k0 = 0; k0 < K; k0 += 4) {
        const int ka = k0 + half * 2;
        v2f a, b;
        a[0] = Arow[ka];
        a[1] = Arow[ka + 1];
        b[0] = B[(size_t)ka       * ldb + n0 + l];
        b[1] = B[(size_t)(ka + 1) * ldb + n0 + l];
        acc = __builtin_amdgcn_wmma_f32_16x16x4_f32(
            false, a, false, b, (short)0, acc, false, false);
    }

    const int col = n0 + l;
    const int rowBase = m0 + half * 8;
#pragma unroll
    for (int r = 0; r < 8; ++r)
        C[(size_t)(rowBase + r) * ldc + col] = acc[r];
}

// ------------------------- attention dot products --------------------------
// a_s[n,h] = <xp[n,h,:], att_src[h,:]> ; same for a_d.  One thread per (n,h).
__global__ void att_kernel(const float* __restrict__ xp,
                           const float* __restrict__ asv,
                           const float* __restrict__ adv,
                           float* __restrict__ a_s, float* __restrict__ a_d,
                           int n, int H, int C)
{
    int t = blockIdx.x * blockDim.x + threadIdx.x;
    if (t >= n * H) return;
    int node = t / H, h = t - node * H;
    const float* row = xp + (size_t)node * H * C + (size_t)h * C;
    const float* sv = asv + (size_t)h * C;
    const float* dv = adv + (size_t)h * C;
    float ss = 0.f, dd = 0.f;
    for (int c = 0; c < C; ++c) {
        float v = row[c];
        ss += v * sv[c];
        dd += v * dv[c];
    }
    a_s[t] = ss;
    a_d[t] = dd;
}

// ------------------------- per-layer init ----------------------------------
__global__ void init_layer(float* __restrict__ m, float* __restrict__ denom,
                           float* __restrict__ agg, int n, int H, int HC)
{
    int t = blockIdx.x * blockDim.x + threadIdx.x;
    if (t < n * HC) agg[t] = 0.f;
    if (t < n * H) { m[t] = -INFINITY; denom[t] = 0.f; }
}

// ------------------------- edge helpers ------------------------------------
__device__ __forceinline__ int edge_src(const int* ei, int E, int e) {
    return (e < E) ? ei[e] : (e - E);          // self loops appended
}
__device__ __forceinline__ int edge_dst(const int* ei, int E, int e) {
    return (e < E) ? ei[E + e] : (e - E);
}
__device__ __forceinline__ float leaky02(float x) {
    return x > 0.f ? x : 0.2f * x;
}
// float atomic max via signed/unsigned int monotonicity trick (all-sign safe)
__device__ __forceinline__ void atomicMaxFloat(float* addr, float val) {
    if (val >= 0.f)
        atomicMax((int*)addr, __float_as_int(val));
    else
        atomicMin((unsigned int*)addr, __float_as_uint(val));
}

// segment max of leaky_relu(a_s[src]+a_d[dst]) over incoming edges of dst
__global__ void edge_max_kernel(const int* __restrict__ ei, int E, int EP,
                                const float* __restrict__ a_s,
                                const float* __restrict__ a_d,
                                float* __restrict__ m, int H)
{
    int t = blockIdx.x * blockDim.x + threadIdx.x;
    if (t >= EP * H) return;
    int e = t / H, h = t - e * H;
    int s = edge_src(ei, E, e), d = edge_dst(ei, E, e);
    float ee = leaky02(a_s[s * H + h] + a_d[d * H + h]);
    atomicMaxFloat(&m[d * H + h], ee);
}

// ex = exp(e - m[dst]) ; denom[dst] += ex
__global__ void edge_sum_kernel(const int* __restrict__ ei, int E, int EP,
                                const float* __restrict__ a_s,
                                const float* __restrict__ a_d,
                                const float* __restrict__ m,
                                float* __restrict__ denom,
                                float* __restrict__ exbuf, int H)
{
    int t = blockIdx.x * blockDim.x + threadIdx.x;
    if (t >= EP * H) return;
    int e = t / H, h = t - e * H;
    int s = edge_src(ei, E, e), d = edge_dst(ei, E, e);
    float ee = leaky02(a_s[s * H + h] + a_d[d * H + h]);
    float mm = m[d * H + h];
    if (!isfinite(mm)) mm = 0.f;               // matches reference where()
    float ex = expf(ee - mm);
    exbuf[(size_t)e * H + h] = ex;
    atomicAdd(&denom[d * H + h], ex);
}

// agg[dst] += alpha * xp[src]; one thread per (edge, 4 channels) -> coalesced
__global__ void edge_agg_kernel(const int* __restrict__ ei, int E, int EP,
                                const float* __restrict__ xp,
                                const float* __restrict__ exbuf,
                                const float* __restrict__ denom,
                                float* __restrict__ agg, int H, int C)
{
    const int HC = H * C;
    const int HC4 = HC >> 2;
    int t = blockIdx.x * blockDim.x + threadIdx.x;
    if (t >= EP * HC4) return;
    int e = t / HC4;
    int g = t - e * HC4;
    int chan = g << 2;
    int h = chan / C;                          // C is a multiple of 4
    int s = edge_src(ei, E, e), d = edge_dst(ei, E, e);
    float alpha = exbuf[(size_t)e * H + h] / (denom[d * H + h] + 1e-16f);
    const float4 v = *(const float4*)(xp + (size_t)s * HC + chan);
    float* dst = agg + (size_t)d * HC + chan;
    atomicAdd(dst + 0, v.x * alpha);
    atomicAdd(dst + 1, v.y * alpha);
    atomicAdd(dst + 2, v.z * alpha);
    atomicAdd(dst + 3, v.w * alpha);
}

// ------------------------- epilogues ---------------------------------------
__global__ void finalize_concat(const float* __restrict__ agg,
                                const float* __restrict__ bias,
                                float* __restrict__ out, int n, int HC)
{
    int t = blockIdx.x * blockDim.x + threadIdx.x;
    if (t >= n * HC) return;
    out[t] = agg[t] + bias[t % HC];
}
__global__ void accum_init(float* __restrict__ out,
                           const float* __restrict__ bias, int n, int Ccol)
{
    int t = blockIdx.x * blockDim.x + threadIdx.x;
    if (t >= n * Ccol) return;
    out[t] = bias[t % Ccol];
}
__global__ void accum_add(float* __restrict__ out,
                          const float* __restrict__ agg, float scale, int total)
{
    int t = blockIdx.x * blockDim.x + threadIdx.x;
    if (t >= total) return;
    out[t] += scale * agg[t];
}

// ------------------------- batch norm --------------------------------------
__global__ void stats_zero(float* __restrict__ stats)
{
    if (threadIdx.x < 256) stats[threadIdx.x] = 0.f;
}
// blockDim.x == Ccol (128): thread c sums channel c over a stride of rows.
__global__ void bn_stats(const float* __restrict__ x, float* __restrict__ stats,
                         int n, int Ccol)
{
    int c = threadIdx.x;
    float s = 0.f, q = 0.f;
    for (int r = blockIdx.x; r < n; r += gridDim.x) {
        float v = x[(size_t)r * Ccol + c];
        s += v;
        q += v * v;
    }
    atomicAdd(&stats[c], s);
    atomicAdd(&stats[Ccol + c], q);
}
__global__ void bn_apply(const float* __restrict__ x,
                         const float* __restrict__ stats,
                         const float* __restrict__ gamma,
                         const float* __restrict__ beta,
                         float* __restrict__ out, int n, int Ccol, int do_elu)
{
    int t = blockIdx.x * blockDim.x + threadIdx.x;
    if (t >= n * Ccol) return;
    int c = t % Ccol;
    float inv_n = 1.f / (float)n;
    float mu = stats[c] * inv_n;
    float var = stats[Ccol + c] * inv_n - mu * mu;
    float y = (x[t] - mu) * rsqrtf(var + EPS_BN) * gamma[c] + beta[c];
    if (do_elu) y = (y > 0.f) ? y : (expf(y) - 1.f);
    out[t] = y;
}

// ------------------------- host orchestration ------------------------------
struct Scratch {
    float *xp, *agg, *hbuf, *a_s, *a_d, *m, *denom, *ex, *stats;
};

static void run_gat_pass(const float* hin, const float* W, int ldb,
                         const float* asv, const float* adv,
                         int N, int E, int EP, const int* ei,
                         int H, int C, const Scratch& s, hipStream_t stream)
{
    const int HC = H * C;
    // projection: xp = hin @ W   ([N,128] @ [128,HC], B slice stride ldb)
    {
        int tiles = (N / 16) * (HC / 16);
        wmma_gemm_f32<<<ceil_div(tiles, 8), 256, 0, stream>>>(
            hin, F_IN, W, ldb, s.xp, HC, N, HC, F_IN);
    }
    att_kernel<<<ceil_div(N * H, 256), 256, 0, stream>>>(
        s.xp, asv, adv, s.a_s, s.a_d, N, H, C);
    init_layer<<<ceil_div(N * HC, 256), 256, 0, stream>>>(
        s.m, s.denom, s.agg, N, H, HC);
    edge_max_kernel<<<ceil_div(EP * H, 256), 256, 0, stream>>>(
        ei, E, EP, s.a_s, s.a_d, s.m, H);
    edge_sum_kernel<<<ceil_div(EP * H, 256), 256, 0, stream>>>(
        ei, E, EP, s.a_s, s.a_d, s.m, s.denom, s.ex, H);
    edge_agg_kernel<<<ceil_div(EP * (HC / 4), 256), 256, 0, stream>>>(
        ei, E, EP, s.xp, s.ex, s.denom, s.agg, H, C);
}

static void run_bn(const float* x, float* out, const float* gamma,
                   const float* beta, int N, int do_elu, const Scratch& s,
                   hipStream_t stream)
{
    stats_zero<<<1, 256, 0, stream>>>(s.stats);
    bn_stats<<<1024, HIDC, 0, stream>>>(x, s.stats, N, HIDC);
    bn_apply<<<ceil_div(N * HIDC, 256), 256, 0, stream>>>(
        x, s.stats, gamma, beta, out, N, HIDC, do_elu);
}

extern "C" void kernel_launch(void* const* d_in, const int* in_sizes, int n_in,
                              void* d_out, int out_size, void* d_ws, size_t ws_size,
                              hipStream_t stream)
{
    const float* x  = (const float*)d_in[0];
    const int*   ei = (const int*)  d_in[1];
    const float* W0  = (const float*)d_in[2];
    const float* as0 = (const float*)d_in[3];
    const float* ad0 = (const float*)d_in[4];
    const float* b0  = (const float*)d_in[5];
    const float* g0  = (const float*)d_in[6];
    const float* be0 = (const float*)d_in[7];
    const float* W1  = (const float*)d_in[8];
    const float* as1 = (const float*)d_in[9];
    const float* ad1 = (const float*)d_in[10];
    const float* b1  = (const float*)d_in[11];
    const float* g1  = (const float*)d_in[12];
    const float* be1 = (const float*)d_in[13];
    const float* W2  = (const float*)d_in[14];
    const float* as2 = (const float*)d_in[15];
    const float* ad2 = (const float*)d_in[16];
    const float* b2  = (const float*)d_in[17];
    const float* g2  = (const float*)d_in[18];
    const float* be2 = (const float*)d_in[19];

    const int N  = in_sizes[0] / F_IN;
    const int E  = in_sizes[1] / 2;
    const int EP = E + N;                       // with self loops

    // workspace layout (all offsets multiples of 256 bytes)
    char* ws = (char*)d_ws;
    size_t nodeF = (size_t)N * HIDC * sizeof(float);   // 25.6 MB
    size_t nodeH = (size_t)N * 4 * sizeof(float);      // 0.8 MB
    size_t off = 0;
    Scratch s;
    s.xp    = (float*)(ws + off); off += nodeF;
    s.agg   = (float*)(ws + off); off += nodeF;
    s.hbuf  = (float*)(ws + off); off += nodeF;
    s.a_s   = (float*)(ws + off); off += nodeH;
    s.a_d   = (float*)(ws + off); off += nodeH;
    s.m     = (float*)(ws + off); off += nodeH;
    s.denom = (float*)(ws + off); off += nodeH;
    s.ex    = (float*)(ws + off); off += (size_t)EP * 4 * sizeof(float);
    s.stats = (float*)(ws + off); off += 256 * sizeof(float);
    (void)ws_size; (void)n_in; (void)out_size;

    float* out = (float*)d_out;

    // ---- layer 0: GAT(128 -> 4x32, concat) + BN + ELU ----
    run_gat_pass(x, W0, HIDC, as0, ad0, N, E, EP, ei, 4, 32, s, stream);
    finalize_concat<<<ceil_div(N * HIDC, 256), 256, 0, stream>>>(
        s.agg, b0, s.hbuf, N, HIDC);
    run_bn(s.hbuf, s.hbuf, g0, be0, N, /*elu=*/1, s, stream);

    // ---- layer 1: GAT(128 -> 4x32, concat) + BN + ELU ----
    run_gat_pass(s.hbuf, W1, HIDC, as1, ad1, N, E, EP, ei, 4, 32, s, stream);
    finalize_concat<<<ceil_div(N * HIDC, 256), 256, 0, stream>>>(
        s.agg, b1, s.hbuf, N, HIDC);
    run_bn(s.hbuf, s.hbuf, g1, be1, N, /*elu=*/1, s, stream);

    // ---- layer 2: GAT(128 -> 4 heads x 128, mean over heads) + BN ----
    // heads are independent through the softmax, so process one head at a
    // time (keeps workspace at [N,128] instead of [N,512]).
    accum_init<<<ceil_div(N * HIDC, 256), 256, 0, stream>>>(out, b2, N, HIDC);
    for (int h = 0; h < 4; ++h) {
        run_gat_pass(s.hbuf, W2 + h * HIDC, 4 * HIDC,
                     as2 + h * HIDC, ad2 + h * HIDC,
                     N, E, EP, ei, 1, HIDC, s, stream);
        accum_add<<<ceil_div(N * HIDC, 256), 256, 0, stream>>>(
            out, s.agg, 0.25f, N * HIDC);
    }
    run_bn(out, out, g2, be2, N, /*elu=*/0, s, stream);
}